// weakSaliencyMechanism_40011915329905
// MI455X (gfx1250) — compile-verified
//
#include <hip/hip_runtime.h>
#include <hip/hip_bf16.h>

typedef float v2f __attribute__((ext_vector_type(2)));
typedef float v4f __attribute__((ext_vector_type(4)));
typedef float v8f __attribute__((ext_vector_type(8)));

// Problem constants
#define NB   32
#define NC   512
#define NHW  3136          // 56*56
#define NHW4 784           // NHW / 4

// ---------------------------------------------------------------------------
// Kernel 1: alpha[b,c] = mean_hw x[b,c,:,:]   (one 256-thread block per (b,c))
// Reads 205 MB once, fully coalesced float4 loads.
// ---------------------------------------------------------------------------
__global__ void k_alpha(const float* __restrict__ x, float* __restrict__ alpha) {
    __shared__ float sbuf[8];
    const int row = blockIdx.x;                       // b*512 + c
    const v4f* xr = reinterpret_cast<const v4f*>(x + (size_t)row * NHW);
    float acc = 0.0f;
    for (int j = threadIdx.x; j < NHW4; j += 256) {
        v4f v = xr[j];
        acc += (v.x + v.y) + (v.z + v.w);
    }
    // wave32 reduce
    #pragma unroll
    for (int off = 16; off > 0; off >>= 1)
        acc += __shfl_down(acc, off, 32);
    const int lane = threadIdx.x & 31;
    const int wave = threadIdx.x >> 5;
    if (lane == 0) sbuf[wave] = acc;
    __syncthreads();
    if (threadIdx.x == 0) {
        float t = 0.0f;
        #pragma unroll
        for (int i = 0; i < 8; ++i) t += sbuf[i];
        alpha[row] = t * (1.0f / (float)NHW);
    }
}

// ---------------------------------------------------------------------------
// Kernel 2: sal[b,hw] = (1/C) * sum_c alpha[b,c] * x[b,c,hw]
// Per-batch matvec mapped onto V_WMMA_F32_16X16X4_F32 (wave32).
//   A[16x4]  : alpha[c0+k] replicated across all rows M
//   B[4x16]  : x[c0+k][hw0+n]  (K on rows, 16 hw positions on columns)
//   D[16x16] : every row == partial sal for the 16 hw columns
// f32 operand layout (ISA §7.12.2): lanes 0-15 carry K=0,1 in VGPR0/1,
// lanes 16-31 carry K=2,3.  Each wave owns one 16-wide hw tile; 512 channels
// = 128 accumulating WMMAs.  Each x element is loaded exactly once overall.
// ---------------------------------------------------------------------------
__global__ void k_sal_wmma(const float* __restrict__ x,
                           const float* __restrict__ alpha,
                           float* __restrict__ sal) {
    const int lane = threadIdx.x & 31;
    const int wave = threadIdx.x >> 5;
    const int tile = blockIdx.x * 4 + wave;           // 0..195 (196*16 = 3136)
    const int b    = blockIdx.y;
    const int hw0  = tile * 16;

    const float* xb = x + (size_t)b * NC * NHW;
    const float* al = alpha + b * NC;

    const int n  = lane & 15;                         // column (hw) index
    const int kh = lane >> 4;                         // 0 -> K=0,1 ; 1 -> K=2,3
    const float* xp = xb + (size_t)(2 * kh) * NHW + hw0 + n;

    v8f acc = {0.f, 0.f, 0.f, 0.f, 0.f, 0.f, 0.f, 0.f};
    for (int c0 = 0; c0 < NC; c0 += 4) {
        const float* p = xp + (size_t)c0 * NHW;
        v2f bmat, amat;
        bmat.x = p[0];                                // x[c0+2*kh  ][hw0+n]
        bmat.y = p[NHW];                              // x[c0+2*kh+1][hw0+n]
        amat.x = al[c0 + 2 * kh];                     // alpha, replicated over M
        amat.y = al[c0 + 2 * kh + 1];
        acc = __builtin_amdgcn_wmma_f32_16x16x4_f32(
            /*neg_a=*/false, amat, /*neg_b=*/false, bmat,
            /*c_mod=*/(short)0, acc, /*reuse_a=*/false, /*reuse_b=*/false);
    }
    // D row M=0 lives in VGPR0 of lanes 0..15 (N = lane); all rows identical.
    if (lane < 16)
        sal[(size_t)b * NHW + hw0 + lane] = acc[0] * (1.0f / (float)NC);
}

// ---------------------------------------------------------------------------
// Kernel 3: out[b,co,hw] = 1 - wf*sigmoid(scale * thresh(relu(sal[b,hw]*w[co])))
// Streaming 205 MB write with non-temporal float4 stores (keep sal in L2).
// relu is subsumed by the >0.01 threshold.
// ---------------------------------------------------------------------------
__global__ void k_out(const float* __restrict__ sal,
                      const float* __restrict__ w,
                      const float* __restrict__ scale,
                      const float* __restrict__ wf,
                      float* __restrict__ out) {
    const size_t i = (size_t)blockIdx.x * blockDim.x + threadIdx.x;  // float4 idx
    const int    hw4 = (int)(i % NHW4);
    const size_t bc  = i / NHW4;
    const int    co  = (int)(bc % NC);
    const int    b   = (int)(bc / NC);

    const float wv  = w[co];
    const float sc  = scale[0];
    const float wfv = wf[0];

    const v4f s4 = reinterpret_cast<const v4f*>(sal + (size_t)b * NHW)[hw4];

    v4f o;
    #pragma unroll
    for (int k = 0; k < 4; ++k) {
        float z  = s4[k] * wv;
        float fm = (z > 0.01f) ? z : 0.0f;            // relu + threshold
        float e  = __expf(-sc * fm);
        o[k] = 1.0f - wfv * (1.0f / (1.0f + e));
    }
    __builtin_nontemporal_store(o, reinterpret_cast<v4f*>(out) + i);
}

// ---------------------------------------------------------------------------
extern "C" void kernel_launch(void* const* d_in, const int* in_sizes, int n_in,
                              void* d_out, int out_size, void* d_ws, size_t ws_size,
                              hipStream_t stream) {
    const float* x      = (const float*)d_in[0];   // [32,512,56,56]
    const float* conv_w = (const float*)d_in[1];   // [512]
    const float* scale  = (const float*)d_in[2];   // [1]
    const float* wf     = (const float*)d_in[3];   // [1]
    float* out = (float*)d_out;

    float* alpha = (float*)d_ws;                               // 16384 f32 = 64 KB
    float* sal   = (float*)((char*)d_ws + NB * NC * sizeof(float)); // 100352 f32

    // Pass 1: per-(b,c) spatial means
    k_alpha<<<NB * NC, 256, 0, stream>>>(x, alpha);

    // Pass 2: channel-weighted reduction via f32 WMMA (4 waves/block, 49*4=196 tiles)
    k_sal_wmma<<<dim3(49, NB), 128, 0, stream>>>(x, alpha, sal);

    // Pass 3: broadcast * conv_w, threshold, sigmoid, streaming store
    const int total4 = NB * NC * NHW4;             // 12,845,056 float4s
    k_out<<<total4 / 256, 256, 0, stream>>>(sal, conv_w, scale, wf, out);
}